// Aggregator_40450001993946
// MI455X (gfx1250) — compile-verified
//
#include <hip/hip_runtime.h>
#include <hip/hip_bf16.h>
#include <math.h>

typedef __attribute__((ext_vector_type(2))) float v2f;
typedef __attribute__((ext_vector_type(8))) float v8f;

#define D       128
#define SECS    5
#define OUTD    128
#define EPS_STD 1e-5f
#define EPS_BN  1e-5f

// ---- sign-aware float atomic min/max (init mn=+INF, mx=-INF) ----
__device__ __forceinline__ void atomicMinF(float* addr, float v) {
    if (v >= 0.0f) atomicMin((int*)addr, __float_as_int(v));
    else           atomicMax((unsigned int*)addr, __float_as_uint(v));
}
__device__ __forceinline__ void atomicMaxF(float* addr, float v) {
    if (v >= 0.0f) atomicMax((int*)addr, __float_as_int(v));
    else           atomicMin((unsigned int*)addr, __float_as_uint(v));
}

// ---- 1) init accumulators ----
__global__ void init_ws(float* __restrict__ sum, float* __restrict__ sumsq,
                        float* __restrict__ mn, float* __restrict__ mx,
                        float* __restrict__ cnt, float* __restrict__ bnacc, int npad) {
    int idx = blockIdx.x * blockDim.x + threadIdx.x;
    int tot = npad * D;
    if (idx < tot) {
        sum[idx]   = 0.0f;
        sumsq[idx] = 0.0f;
        mn[idx]    =  __builtin_inff();
        mx[idx]    = -__builtin_inff();
    }
    if (idx < npad)   cnt[idx]   = 0.0f;
    if (idx < 2*OUTD) bnacc[idx] = 0.0f;
}

// ---- 2) segment aggregation via L2 atomics ----
__global__ void __launch_bounds__(256)
aggregate(const float* __restrict__ inputs, const int* __restrict__ index,
          float* __restrict__ sum, float* __restrict__ sumsq,
          float* __restrict__ mn, float* __restrict__ mx,
          float* __restrict__ cnt, int E) {
    const int d   = threadIdx.x & (D - 1);
    const int sub = threadIdx.x >> 7;            // 0 or 1
    int base = blockIdx.x * 16 + sub * 8;
#pragma unroll
    for (int k = 0; k < 8; ++k) {
        int e = base + k;
        if (e >= E) return;
        int n = index[e];
        float v = inputs[(size_t)e * D + d];
        size_t o = (size_t)n * D + d;
        atomicAdd(sum + o, v);
        atomicAdd(sumsq + o, v * v);
        atomicMinF(mn + o, v);
        atomicMaxF(mx + o, v);
        if (d == 0) atomicAdd(cnt + n, 1.0f);
    }
}

// ---- 3) finalize: sum->mean, sumsq->std, fix empty min/max, degree index ----
__global__ void finalize(float* __restrict__ sum, float* __restrict__ sumsq,
                         float* __restrict__ mn, float* __restrict__ mx,
                         const float* __restrict__ cnt, int* __restrict__ degidx, int npad) {
    int idx = blockIdx.x * blockDim.x + threadIdx.x;
    if (idx >= npad * D) return;
    int n = idx >> 7;
    float c    = cnt[n];
    float safe = fmaxf(c, 1.0f);
    float m  = sum[idx]   / safe;
    float ms = sumsq[idx] / safe;
    sum[idx]   = m;
    sumsq[idx] = sqrtf(fmaxf(ms - m * m, 0.0f) + EPS_STD);   // std (empty -> sqrt(eps))
    bool has = c > 0.0f;
    float a = mn[idx], b = mx[idx];
    mn[idx] = has ? a : 0.0f;
    mx[idx] = has ? b : 0.0f;
    if ((idx & (D - 1)) == 0) {
        int dg = (int)c;
        if (dg > 99) dg = 99;
        if (dg < 0)  dg = 0;
        degidx[n] = dg;
    }
}

// ---- 4) [Npad x 640] x [640 x 128] GEMM via V_WMMA_F32_16X16X4_F32 ----
// One wave32 per 16x16 output tile. A-layout (f32 16x4): lane L holds
// A[L%16][(L/16)*2 + {0,1}]; B-layout mirrored; C/D: VGPR r holds M = r + 8*(L/16), N = L%16.
__global__ void __launch_bounds__(256)
gemm_wmma(const float* __restrict__ mean, const float* __restrict__ mn,
          const float* __restrict__ mx,   const float* __restrict__ stdv,
          const int* __restrict__ degidx, const float* __restrict__ demb,
          const float* __restrict__ Wm,   float* __restrict__ out, int N) {
    const int wave = threadIdx.x >> 5;
    const int lane = threadIdx.x & 31;
    const int tile = blockIdx.x * 8 + wave;      // 8 waves per block, 8 col-tiles per row-tile
    const int rowTile = tile >> 3;
    const int colTile = tile & 7;
    const int lmod  = lane & 15;
    const int lhi   = lane >> 4;                 // 0 or 1
    const int khalf = lhi * 2;                   // K sub-offset for this half-wave
    const int row = rowTile * 16 + lmod;
    const int col = colTile * 16 + lmod;

    const float* aptr[SECS];
    aptr[0] = mean + (size_t)row * D;
    aptr[1] = mn   + (size_t)row * D;
    aptr[2] = mx   + (size_t)row * D;
    aptr[3] = stdv + (size_t)row * D;
    aptr[4] = demb + (size_t)degidx[row] * D;

    v8f c = {};
#pragma unroll
    for (int sec = 0; sec < SECS; ++sec) {
        const float* ap = aptr[sec] + khalf;
        const float* wp = Wm + (size_t)(sec * D + khalf) * OUTD + col;
#pragma unroll 4
        for (int k = 0; k < D; k += 4) {
            v2f a; a.x = ap[k];            a.y = ap[k + 1];
            v2f b; b.x = wp[k * OUTD];     b.y = wp[k * OUTD + OUTD];
            c = __builtin_amdgcn_wmma_f32_16x16x4_f32(
                    false, a, false, b, (short)0, c, false, false);
        }
    }
    const int mbase = rowTile * 16 + lhi * 8;
#pragma unroll
    for (int r = 0; r < 8; ++r) {
        int m = mbase + r;
        if (m < N) out[(size_t)m * OUTD + col] = c[r];
    }
}

// ---- 5) BatchNorm column statistics ----
__global__ void __launch_bounds__(128)
bn_stats(const float* __restrict__ h, float* __restrict__ bnacc, int N) {
    const int col = threadIdx.x;                 // 128 threads, coalesced row reads
    float s = 0.0f, q = 0.0f;
    for (int n = blockIdx.x; n < N; n += gridDim.x) {
        float v = h[(size_t)n * OUTD + col];
        s += v;
        q += v * v;
    }
    atomicAdd(&bnacc[col], s);
    atomicAdd(&bnacc[OUTD + col], q);
}

// ---- 6) BatchNorm apply + ReLU (in place on d_out) ----
__global__ void bn_apply(float* __restrict__ h, const float* __restrict__ bnacc,
                         const float* __restrict__ gamma, const float* __restrict__ beta, int N) {
    int idx = blockIdx.x * blockDim.x + threadIdx.x;
    if (idx >= N * OUTD) return;
    int j = idx & (OUTD - 1);
    float invN = 1.0f / (float)N;
    float mu  = bnacc[j] * invN;
    float var = fmaxf(bnacc[OUTD + j] * invN - mu * mu, 0.0f);
    float scale = gamma[j] * rsqrtf(var + EPS_BN);
    float shift = beta[j] - mu * scale;
    float v = h[idx] * scale + shift;
    h[idx] = fmaxf(v, 0.0f);
}

extern "C" void kernel_launch(void* const* d_in, const int* in_sizes, int n_in,
                              void* d_out, int out_size, void* d_ws, size_t ws_size,
                              hipStream_t stream) {
    (void)n_in; (void)ws_size;
    const float* inputs = (const float*)d_in[0];
    const int*   index  = (const int*)  d_in[1];
    const float* demb   = (const float*)d_in[2];
    const float* Wm     = (const float*)d_in[3];
    const float* gamma  = (const float*)d_in[4];
    const float* beta   = (const float*)d_in[5];
    // d_in[6] is dim_size on device; derive host-side instead:
    const int E = in_sizes[0] / D;
    const int N = out_size / OUTD;
    const int Npad = (N + 15) & ~15;
    const int rowTiles = Npad / 16;

    // workspace layout (floats): sum | sumsq | min | max | cnt | degidx | bnacc
    float* ws    = (float*)d_ws;
    size_t nd    = (size_t)Npad * D;
    float* sum   = ws;
    float* sumsq = sum + nd;
    float* mnb   = sumsq + nd;
    float* mxb   = mnb + nd;
    float* cnt   = mxb + nd;
    int*   degix = (int*)(cnt + Npad);
    float* bnacc = (float*)(degix + Npad);

    {   // init
        int tot = Npad * D;
        init_ws<<<(tot + 255) / 256, 256, 0, stream>>>(sum, sumsq, mnb, mxb, cnt, bnacc, Npad);
    }
    {   // aggregate: 16 edges per 256-thread block
        int bl = (E + 15) / 16;
        aggregate<<<bl, 256, 0, stream>>>(inputs, index, sum, sumsq, mnb, mxb, cnt, E);
    }
    {   // finalize
        int tot = Npad * D;
        finalize<<<(tot + 255) / 256, 256, 0, stream>>>(sum, sumsq, mnb, mxb, cnt, degix, Npad);
    }
    // GEMM: 8 waves/block, one block per row-tile (covers all 8 col-tiles)
    gemm_wmma<<<rowTiles, 256, 0, stream>>>(sum, mnb, mxb, sumsq, degix, demb, Wm,
                                            (float*)d_out, N);
    // BatchNorm
    bn_stats<<<512, 128, 0, stream>>>((const float*)d_out, bnacc, N);
    bn_apply<<<(N * OUTD + 255) / 256, 256, 0, stream>>>((float*)d_out, bnacc, gamma, beta, N);
}